// GCN_6227702579493
// MI455X (gfx1250) — compile-verified
//
#include <hip/hip_runtime.h>
#include <math.h>

#define NNODES 50000
#define NEDGES 800000
#define FEATS  128
#define NGRAPH 64

typedef __attribute__((ext_vector_type(2))) float v2f;
typedef __attribute__((ext_vector_type(8))) float v8f;

// ---------------------------------------------------------------- utilities
__global__ __launch_bounds__(256) void zero_f32(float* __restrict__ p, int n) {
  int i = blockIdx.x * 256 + threadIdx.x;
  if (i < n) p[i] = 0.0f;
}

__device__ __forceinline__ int lower_bound_i(const int* __restrict__ a, int n, int key) {
  int lo = 0, hi = n;
  while (lo < hi) { int mid = (lo + hi) >> 1; if (a[mid] < key) lo = mid + 1; else hi = mid; }
  return lo;
}

// ---------------------------------------------------------------- degrees
__global__ __launch_bounds__(256) void degree_kernel(const int* __restrict__ src,
                                                     const int* __restrict__ dst,
                                                     int* __restrict__ deg_out,
                                                     int* __restrict__ deg_in) {
  int e = blockIdx.x * 256 + threadIdx.x;
  if (e < NEDGES) {
    atomicAdd(&deg_out[src[e]], 1);
    atomicAdd(&deg_in[dst[e]], 1);
  }
}

__global__ __launch_bounds__(256) void invsqrt_kernel(const int* __restrict__ deg_out,
                                                      const int* __restrict__ deg_in,
                                                      float* __restrict__ inv_out,
                                                      float* __restrict__ inv_in) {
  int i = blockIdx.x * 256 + threadIdx.x;
  if (i < NNODES) {
    inv_out[i] = rsqrtf(fmaxf((float)deg_out[i], 1.0f));
    inv_in[i]  = rsqrtf(fmaxf((float)deg_in[i], 1.0f));
  }
}

// ---------------------------------------------------------------- WMMA GEMM
// Y[row, n] = sum_k (X[row,k] * scale[row]) * W[k,n]
// One wave computes a 16x16 tile via V_WMMA_F32_16X16X4_F32 (native fp32).
// Block = 8 waves = all 128 output columns for 16 node rows. Grid = NNODES/16.
__global__ __launch_bounds__(256) void gemm128_wmma(const float* __restrict__ X,
                                                    const float* __restrict__ scale,
                                                    const float* __restrict__ W,
                                                    float* __restrict__ Y) {
  const int lane = threadIdx.x & 31;
  const int wave = threadIdx.x >> 5;
  const int lo = lane & 15;          // M (for A/C) or N (for B/C) index
  const int hi = lane >> 4;          // K-half selector
  const int row0 = blockIdx.x * 16;
  const int row = row0 + lo;
  const int n0 = wave * 16;
  const float s = scale[row];
  const float* xr = X + (size_t)row * FEATS;

  v8f c = {};
#pragma unroll
  for (int k = 0; k < FEATS; k += 4) {
    // A 16x4: vgpr0 = K=k+2*hi, vgpr1 = K=k+2*hi+1, row = lo  (aligned b64 load)
    v2f a = *(const v2f*)(xr + k + 2 * hi);
    a.x *= s; a.y *= s;
    // B 4x16: vgpr0 lanes hold K=k+2*hi at col n0+lo, vgpr1 holds K=k+2*hi+1
    v2f b;
    b.x = W[(size_t)(k + 2 * hi) * FEATS + n0 + lo];
    b.y = W[(size_t)(k + 2 * hi + 1) * FEATS + n0 + lo];
    c = __builtin_amdgcn_wmma_f32_16x16x4_f32(false, a, false, b, (short)0, c,
                                              false, false);
  }
  // C 16x16 f32: vgpr j -> M = j + 8*hi, N = lo
#pragma unroll
  for (int j = 0; j < 8; ++j)
    Y[(size_t)(row0 + j + 8 * hi) * FEATS + n0 + lo] = c[j];
}

// ---------------------------------------------------------------- edge scatter
// One wave per edge; lane handles 4 features (float4 gather + 4 f32 atomics).
__global__ __launch_bounds__(256) void scatter_add(const float* __restrict__ P,
                                                   const int* __restrict__ src,
                                                   const int* __restrict__ dst,
                                                   float* __restrict__ agg) {
  int e = (int)(((unsigned)(blockIdx.x * 256 + threadIdx.x)) >> 5);
  int lane = threadIdx.x & 31;
  const float4 v = *(const float4*)(P + (size_t)src[e] * FEATS + lane * 4);
  float* o = agg + (size_t)dst[e] * FEATS + lane * 4;
  __hip_atomic_fetch_add(o + 0, v.x, __ATOMIC_RELAXED, __HIP_MEMORY_SCOPE_AGENT);
  __hip_atomic_fetch_add(o + 1, v.y, __ATOMIC_RELAXED, __HIP_MEMORY_SCOPE_AGENT);
  __hip_atomic_fetch_add(o + 2, v.z, __ATOMIC_RELAXED, __HIP_MEMORY_SCOPE_AGENT);
  __hip_atomic_fetch_add(o + 3, v.w, __ATOMIC_RELAXED, __HIP_MEMORY_SCOPE_AGENT);
}

// ---------------------------------------------------------------- relu(agg*inv_in + b), re-zero agg
__global__ __launch_bounds__(256) void post_relu(float* __restrict__ agg,
                                                 const float* __restrict__ inv_in,
                                                 const float* __restrict__ bias,
                                                 float* __restrict__ h) {
  size_t i = (size_t)blockIdx.x * 256 + threadIdx.x;
  int row = (int)(i >> 7);
  int col = (int)(i & 127);
  float v = agg[i] * inv_in[row] + bias[col];
  h[i] = v > 0.0f ? v : 0.0f;
  agg[i] = 0.0f;  // ready for next scatter pass
}

// ---------------------------------------------------------------- gate = h @ gate_W + gate_b
__global__ __launch_bounds__(256) void gate_kernel(const float* __restrict__ h,
                                                   const float* __restrict__ gw,
                                                   const float* __restrict__ gb,
                                                   float* __restrict__ gate) {
  int node = (int)(((unsigned)(blockIdx.x * 256 + threadIdx.x)) >> 5);
  int lane = threadIdx.x & 31;
  const float4 hv = *(const float4*)(h + (size_t)node * FEATS + lane * 4);
  const float4 wv = *(const float4*)(gw + lane * 4);
  float p = hv.x * wv.x + hv.y * wv.y + hv.z * wv.z + hv.w * wv.w;
  p += __shfl_xor(p, 16);
  p += __shfl_xor(p, 8);
  p += __shfl_xor(p, 4);
  p += __shfl_xor(p, 2);
  p += __shfl_xor(p, 1);
  if (lane == 0) gate[node] = p + gb[0];
}

// ---------------------------------------------------------------- per-graph max & sum(exp)
__global__ __launch_bounds__(256) void softmax_stats(const int* __restrict__ gid,
                                                     const float* __restrict__ gate,
                                                     float* __restrict__ gmax,
                                                     float* __restrict__ denom) {
  int g = blockIdx.x;
  int lo = lower_bound_i(gid, NNODES, g);
  int hi = lower_bound_i(gid, NNODES, g + 1);
  __shared__ float red[256];
  float m = -3.4e38f;
  for (int i = lo + threadIdx.x; i < hi; i += 256) m = fmaxf(m, gate[i]);
  red[threadIdx.x] = m;
  __syncthreads();
  for (int t = 128; t > 0; t >>= 1) {
    if (threadIdx.x < t) red[threadIdx.x] = fmaxf(red[threadIdx.x], red[threadIdx.x + t]);
    __syncthreads();
  }
  float gm = red[0];
  __syncthreads();
  float sum = 0.0f;
  for (int i = lo + threadIdx.x; i < hi; i += 256) sum += expf(gate[i] - gm);
  red[threadIdx.x] = sum;
  __syncthreads();
  for (int t = 128; t > 0; t >>= 1) {
    if (threadIdx.x < t) red[threadIdx.x] += red[threadIdx.x + t];
    __syncthreads();
  }
  if (threadIdx.x == 0) {
    if (hi == lo) { gmax[g] = 0.0f; denom[g] = 1.0f; }  // empty graph -> alpha sums to 0
    else          { gmax[g] = gm;   denom[g] = red[0]; }
  }
}

__global__ __launch_bounds__(256) void alpha_kernel(const int* __restrict__ gid,
                                                    const float* __restrict__ gate,
                                                    const float* __restrict__ gmax,
                                                    const float* __restrict__ denom,
                                                    float* __restrict__ alpha) {
  int i = blockIdx.x * 256 + threadIdx.x;
  if (i < NNODES) {
    int g = gid[i];
    alpha[i] = expf(gate[i] - gmax[g]) / denom[g];
  }
}

// ---------------------------------------------------------------- attention pool + final linear
__global__ __launch_bounds__(128) void pool_out(const int* __restrict__ gid,
                                                const float* __restrict__ alpha,
                                                const float* __restrict__ h,
                                                const float* __restrict__ ow,
                                                const float* __restrict__ ob,
                                                float* __restrict__ out) {
  int g = blockIdx.x;
  int f = threadIdx.x;  // 0..127
  int lo = lower_bound_i(gid, NNODES, g);
  int hi = lower_bound_i(gid, NNODES, g + 1);
  float s = 0.0f;
  for (int n = lo; n < hi; ++n) s += alpha[n] * h[(size_t)n * FEATS + f];
  __shared__ float red[128];
  red[f] = s * ow[f];
  __syncthreads();
  for (int t = 64; t > 0; t >>= 1) {
    if (f < t) red[f] += red[f + t];
    __syncthreads();
  }
  if (f == 0) out[g] = red[0] + ob[0];
}

// ---------------------------------------------------------------- launcher
extern "C" void kernel_launch(void* const* d_in, const int* in_sizes, int n_in,
                              void* d_out, int out_size, void* d_ws, size_t ws_size,
                              hipStream_t stream) {
  const float* x      = (const float*)d_in[0];
  const float* W1     = (const float*)d_in[1];
  const float* b1     = (const float*)d_in[2];
  const float* W2     = (const float*)d_in[3];
  const float* b2     = (const float*)d_in[4];
  const float* gate_W = (const float*)d_in[5];
  const float* gate_b = (const float*)d_in[6];
  const float* out_W  = (const float*)d_in[7];
  const float* out_b  = (const float*)d_in[8];
  const int*   src    = (const int*)d_in[9];
  const int*   dst    = (const int*)d_in[10];
  const int*   gid    = (const int*)d_in[11];
  float* out = (float*)d_out;

  float* ws      = (float*)d_ws;
  float* proj    = ws;                                   // NNODES*FEATS
  float* agg     = proj    + (size_t)NNODES * FEATS;     // NNODES*FEATS
  float* h       = agg     + (size_t)NNODES * FEATS;     // NNODES*FEATS
  float* inv_out = h       + (size_t)NNODES * FEATS;     // NNODES
  float* inv_in  = inv_out + NNODES;                     // NNODES
  float* gate    = inv_in  + NNODES;                     // NNODES
  float* alpha   = gate    + NNODES;                     // NNODES
  float* gmax    = alpha   + NNODES;                     // NGRAPH
  float* denom   = gmax    + NGRAPH;                     // NGRAPH
  int*   deg_out = (int*)(denom + NGRAPH);               // NNODES
  int*   deg_in  = deg_out + NNODES;                     // NNODES

  const int NF = NNODES * FEATS;  // 6,400,000 (multiple of 256)

  // Re-initialize scratch every call (deterministic, no cross-call state).
  zero_f32<<<NF / 256, 256, 0, stream>>>(agg, NF);
  zero_f32<<<(2 * NNODES + 255) / 256, 256, 0, stream>>>((float*)deg_out, 2 * NNODES);

  degree_kernel<<<NEDGES / 256, 256, 0, stream>>>(src, dst, deg_out, deg_in);
  invsqrt_kernel<<<(NNODES + 255) / 256, 256, 0, stream>>>(deg_out, deg_in, inv_out, inv_in);

  // Layer 1: proj = (x * inv_out) @ W1 ; agg = scatter ; h = relu(agg*inv_in + b1)
  gemm128_wmma<<<NNODES / 16, 256, 0, stream>>>(x, inv_out, W1, proj);
  scatter_add<<<NEDGES / 8, 256, 0, stream>>>(proj, src, dst, agg);
  post_relu<<<NF / 256, 256, 0, stream>>>(agg, inv_in, b1, h);

  // Layer 2 (agg re-zeroed by post_relu above)
  gemm128_wmma<<<NNODES / 16, 256, 0, stream>>>(h, inv_out, W2, proj);
  scatter_add<<<NEDGES / 8, 256, 0, stream>>>(proj, src, dst, agg);
  post_relu<<<NF / 256, 256, 0, stream>>>(agg, inv_in, b2, h);

  // Gated attention pooling + output head
  gate_kernel<<<NNODES / 8, 256, 0, stream>>>(h, gate_W, gate_b, gate);
  softmax_stats<<<NGRAPH, 256, 0, stream>>>(gid, gate, gmax, denom);
  alpha_kernel<<<(NNODES + 255) / 256, 256, 0, stream>>>(gid, gate, gmax, denom, alpha);
  pool_out<<<NGRAPH, 128, 0, stream>>>(gid, alpha, h, out_W, out_b, out);
}